// RoseTTAFoldBlock_19413252178131
// MI455X (gfx1250) — compile-verified
//
#include <hip/hip_runtime.h>
#include <cstddef>
#include <cstdint>

// ---------------------------------------------------------------------------
// Shapes (fixed by the reference)
// ---------------------------------------------------------------------------
#define NN_    1024
#define D1_    128
#define D2_    64
#define H_     4
#define C_     32

typedef __attribute__((ext_vector_type(2))) float v2f;
typedef __attribute__((ext_vector_type(8))) float v8f;
typedef __attribute__((ext_vector_type(4))) int   v4i;

// CDNA5 fp32 WMMA: D(16x16,f32) = A(16x4,f32) x B(4x16,f32) + C
__device__ __forceinline__ v8f wmma4(v2f a, v2f b, v8f c) {
  return __builtin_amdgcn_wmma_f32_16x16x4_f32(
      /*neg_a=*/false, a, /*neg_b=*/false, b,
      /*c_mod=*/(short)0, c, /*reuse_a=*/false, /*reuse_b=*/false);
}

// ---------------------------------------------------------------------------
// CDNA5 async global->LDS copy (ASYNCcnt path). 16B per lane per issue.
// Builtin signature (from hipcc diagnostic): (v4i AS1*, v4i AS3*, imm, imm).
// ---------------------------------------------------------------------------
__device__ __forceinline__ void async_load_b128(void* lds, const void* gptr) {
#if __has_builtin(__builtin_amdgcn_global_load_async_to_lds_b128)
  __builtin_amdgcn_global_load_async_to_lds_b128(
      (__attribute__((address_space(1))) v4i*)gptr,
      (__attribute__((address_space(3))) v4i*)lds,
      /*offset=*/0, /*cpol=*/0);
#else
  const uint32_t loff =
      (uint32_t)(uintptr_t)(__attribute__((address_space(3))) void*)lds;
  asm volatile("global_load_async_to_lds_b128 %0, %1, off"
               :: "v"(loff), "v"(gptr) : "memory");
#endif
}

__device__ __forceinline__ void async_wait0() {
#if __has_builtin(__builtin_amdgcn_s_wait_asynccnt)
  __builtin_amdgcn_s_wait_asynccnt(0);
#else
  asm volatile("s_wait_asynccnt 0" ::: "memory");
#endif
}

// ---------------------------------------------------------------------------
// Generic fp32 WMMA GEMM: one wave (32 lanes) per 16x16 output tile.
//   C[m,n] = act( alpha * sum_k A[m,k]*B(k,n) + bias[n] + resid[m,n] )
// BLAYOUT 0: B stored K x N row-major (B[k*ldb+n])
// BLAYOUT 1: B stored N x K row-major (B[n*ldb+k])  -- used for Q @ K^T
// ---------------------------------------------------------------------------
template <int BLAYOUT, bool GELU>
__global__ __launch_bounds__(32)
void gemm_wmma_f32(const float* __restrict__ A, int lda,
                   const float* __restrict__ B, int ldb,
                   float* __restrict__ C, int ldc, int K,
                   const float* __restrict__ bias,
                   const float* __restrict__ resid, int ldres,
                   float alpha) {
  const int lane = threadIdx.x;
  const int half = lane >> 4;
  const int l16  = lane & 15;
  const int m0 = blockIdx.y * 16;
  const int n0 = blockIdx.x * 16;
  const int arow = m0 + l16;
  const int bcol = n0 + l16;
  v8f acc = {};
  for (int k0 = 0; k0 < K; k0 += 4) {
    const int ka = k0 + 2 * half;
    v2f a, b;
    const float2 av = *reinterpret_cast<const float2*>(A + (size_t)arow * lda + ka);
    a.x = av.x; a.y = av.y;
    if (BLAYOUT == 0) {
      b.x = B[(size_t)ka * ldb + bcol];
      b.y = B[(size_t)(ka + 1) * ldb + bcol];
    } else {
      const float2 bv = *reinterpret_cast<const float2*>(B + (size_t)bcol * ldb + ka);
      b.x = bv.x; b.y = bv.y;
    }
    acc = wmma4(a, b, acc);
  }
#pragma unroll
  for (int r = 0; r < 8; ++r) {
    const int m = m0 + r + 8 * half;
    const int n = n0 + l16;
    float v = acc[r] * alpha;
    if (bias)  v += bias[n];
    if (resid) v += resid[(size_t)m * ldres + n];
    if (GELU)  v = 0.5f * v * (1.0f + erff(v * 0.70710678118f));
    C[(size_t)m * ldc + n] = v;
  }
}

// ---------------------------------------------------------------------------
// Split-K fp32 WMMA GEMM for long-K / low-tile-count cases (att @ V, K=1024):
// 256-thread block = 8 waves per 16x16 output tile; wave w accumulates the
// K/8 chunk with WMMA, then an LDS combine reduces the 8 partials.
// ---------------------------------------------------------------------------
__global__ __launch_bounds__(256)
void gemm_wmma_f32_splitk(const float* __restrict__ A, int lda,
                          const float* __restrict__ B, int ldb,
                          float* __restrict__ C, int ldc, int K,
                          const float* __restrict__ bias,
                          const float* __restrict__ resid, int ldres,
                          float alpha) {
  const int t = threadIdx.x;
  const int wave = t >> 5, lane = t & 31, half = lane >> 4, l16 = lane & 15;
  const int m0 = blockIdx.y * 16;
  const int n0 = blockIdx.x * 16;
  const int arow = m0 + l16;
  const int bcol = n0 + l16;
  const int kchunk = K >> 3;              // K multiple of 32
  const int kbeg = wave * kchunk;
  const int kend = kbeg + kchunk;
  v8f acc = {};
  for (int k0 = kbeg; k0 < kend; k0 += 4) {
    const int ka = k0 + 2 * half;
    v2f a, b;
    const float2 av = *reinterpret_cast<const float2*>(A + (size_t)arow * lda + ka);
    a.x = av.x; a.y = av.y;
    b.x = B[(size_t)ka * ldb + bcol];
    b.y = B[(size_t)(ka + 1) * ldb + bcol];
    acc = wmma4(a, b, acc);
  }
  __shared__ float red[8 * 256];
#pragma unroll
  for (int r = 0; r < 8; ++r) {
    const int ml = r + 8 * half;                 // local m in 0..15
    red[wave * 256 + ml * 16 + l16] = acc[r];
  }
  __syncthreads();
  // one output element per thread
  const int ml = t >> 4, nl = t & 15;
  float v = 0.0f;
#pragma unroll
  for (int w8 = 0; w8 < 8; ++w8) v += red[w8 * 256 + t];
  v *= alpha;
  const int m = m0 + ml, n = n0 + nl;
  if (bias)  v += bias[n];
  if (resid) v += resid[(size_t)m * ldres + n];
  C[(size_t)m * ldc + n] = v;
}

// ---------------------------------------------------------------------------
// Row LayerNorm over D=128, optional residual add on the output.
// ---------------------------------------------------------------------------
__global__ __launch_bounds__(128)
void layernorm128(const float* __restrict__ x, const float* __restrict__ g,
                  const float* __restrict__ b, const float* __restrict__ resid,
                  float* __restrict__ y) {
  const int row = blockIdx.x, t = threadIdx.x;
  const float v = x[(size_t)row * 128 + t];
  __shared__ float red[128];
  red[t] = v; __syncthreads();
  for (int off = 64; off; off >>= 1) { if (t < off) red[t] += red[t + off]; __syncthreads(); }
  const float mean = red[0] * (1.0f / 128.0f); __syncthreads();
  const float d = v - mean;
  red[t] = d * d; __syncthreads();
  for (int off = 64; off; off >>= 1) { if (t < off) red[t] += red[t + off]; __syncthreads(); }
  const float var = red[0] * (1.0f / 128.0f);
  float out = d * rsqrtf(var + 1e-5f) * g[t] + b[t];
  if (resid) out += resid[(size_t)row * 128 + t];
  y[(size_t)row * 128 + t] = out;
}

// ---------------------------------------------------------------------------
// pair_2d.mean(axis=2): (1,N,N,64) -> (N,64). One block per i. Streams 256 MB;
// prefetch a few cachelines ahead (lowers to global_prefetch_b8).
// ---------------------------------------------------------------------------
__global__ __launch_bounds__(256)
void pair_mean(const float* __restrict__ pair, float* __restrict__ out) {
  const int i = blockIdx.x;
  const int c  = threadIdx.x & 63;
  const int jg = threadIdx.x >> 6;   // 4 j-groups
  float s = 0.0f;
  const size_t base = (size_t)i * NN_ * D2_;
  for (int j = jg; j < NN_; j += 4) {
    if (((j - jg) & 31) == 0 && j + 32 < NN_)
      __builtin_prefetch(&pair[base + (size_t)(j + 32) * D2_ + c], 0, 0);
    s += pair[base + (size_t)j * D2_ + c];
  }
  __shared__ float red[256];
  red[threadIdx.x] = s; __syncthreads();
  if (jg == 0)
    out[(size_t)i * D2_ + c] =
        (red[c] + red[64 + c] + red[128 + c] + red[192 + c]) * (1.0f / 1024.0f);
}

// ---------------------------------------------------------------------------
// Row softmax over N=1024 with optional per-column bias (per head group).
// ---------------------------------------------------------------------------
__global__ __launch_bounds__(256)
void softmax_bias(float* __restrict__ logits, const float* __restrict__ bias) {
  const int row = blockIdx.x;
  float* p = logits + (size_t)row * NN_;
  const float* bb = bias ? bias + (size_t)(row >> 10) * NN_ : nullptr;
  const int t = threadIdx.x;
  __shared__ float red[256];
  float vals[4];
  float vmax = -1e30f;
#pragma unroll
  for (int s = 0; s < 4; ++s) {
    const int j = t + s * 256;
    float v = p[j];
    if (bb) v += bb[j];
    vals[s] = v;
    vmax = fmaxf(vmax, v);
  }
  red[t] = vmax; __syncthreads();
  for (int off = 128; off; off >>= 1) { if (t < off) red[t] = fmaxf(red[t], red[t + off]); __syncthreads(); }
  const float m = red[0]; __syncthreads();
  float ssum = 0.0f;
#pragma unroll
  for (int s = 0; s < 4; ++s) { vals[s] = __expf(vals[s] - m); ssum += vals[s]; }
  red[t] = ssum; __syncthreads();
  for (int off = 128; off; off >>= 1) { if (t < off) red[t] += red[t + off]; __syncthreads(); }
  const float inv = 1.0f / red[0];
#pragma unroll
  for (int s = 0; s < 4; ++s) p[t + s * 256] = vals[s] * inv;
}

// ---------------------------------------------------------------------------
// Fused pair path, streamed once:
//   P = pair_2d + pad(left_i + right_j)   ->   out = P + P @ w_conv + b_conv
// Block: 256 thr = 8 waves; tile = (i fixed) x 32 j-rows x 64 channels.
// w_conv (16 KB) and the P tile (8 KB) staged with ASYNC global->LDS copies
// (no VGPR round-trip), outer-sum applied in LDS, then fp32 WMMA over K=64.
// ---------------------------------------------------------------------------
__global__ __launch_bounds__(256)
void pair_conv(const float* __restrict__ pair, const float* __restrict__ left,
               const float* __restrict__ right, const float* __restrict__ wconv,
               const float* __restrict__ bconv, float* __restrict__ out) {
  __shared__ __align__(16) float Wl[64 * 64];
  __shared__ __align__(16) float P[32 * 64];
  const int i  = blockIdx.y;
  const int j0 = blockIdx.x * 32;
  const int t  = threadIdx.x;
  const size_t base = ((size_t)i * NN_ + j0) * D2_;
  // Async stage: w_conv = 1024 x float4, pair tile = 512 x float4.
  for (int idx = t; idx < 1024; idx += 256)
    async_load_b128(&Wl[idx * 4], wconv + (size_t)idx * 4);
  for (int idx = t; idx < 512; idx += 256)
    async_load_b128(&P[idx * 4], pair + base + (size_t)idx * 4);
  async_wait0();
  __syncthreads();
  // outer-sum on first 32 channels (32 rows x 32 chans = 1024 elements)
  for (int idx = t; idx < 1024; idx += 256) {
    const int jj = idx >> 5, c = idx & 31;
    P[jj * 64 + c] += left[(size_t)i * 32 + c] + right[(size_t)(j0 + jj) * 32 + c];
  }
  __syncthreads();
  const int wave = t >> 5, lane = t & 31, half = lane >> 4, l16 = lane & 15;
  const int rbase = (wave >> 2) * 16;   // j sub-tile: 0 or 16
  const int cbase = (wave & 3) * 16;    // channel tile: 0,16,32,48
  v8f acc = {};
  for (int k0 = 0; k0 < 64; k0 += 4) {
    const int ka = k0 + 2 * half;
    v2f a, b;
    a.x = P[(rbase + l16) * 64 + ka];
    a.y = P[(rbase + l16) * 64 + ka + 1];
    b.x = Wl[ka * 64 + cbase + l16];
    b.y = Wl[(ka + 1) * 64 + cbase + l16];
    acc = wmma4(a, b, acc);
  }
#pragma unroll
  for (int r = 0; r < 8; ++r) {
    const int jj = rbase + r + 8 * half;
    const int c  = cbase + l16;
    out[base + (size_t)jj * 64 + c] = acc[r] + bconv[c] + P[jj * 64 + c];
  }
}

// ---------------------------------------------------------------------------
// IPA prep: per-head concatenated score/value features.
// qcat[h][n] = [ q*(wL/sqrt(C)) , qpg*s2h , 0x4 ]   (48 cols)
// kcat[h][n] = [ k , kpg , 0x4 ];  vcat[h][n] = [ v , vpg , 0x4 ]
// kkbias[h][n] = -0.5*s2h*|kpg|^2  (qq term is softmax-invariant -> dropped)
// ---------------------------------------------------------------------------
__global__ __launch_bounds__(128)
void ipa_pointprep(const float* __restrict__ qb, const float* __restrict__ kb,
                   const float* __restrict__ vb, const float* __restrict__ qp,
                   const float* __restrict__ kp, const float* __restrict__ vp,
                   const float* __restrict__ R, const float* __restrict__ tvec,
                   const float* __restrict__ headw,
                   float* __restrict__ qcat, float* __restrict__ kcat,
                   float* __restrict__ vcat, float* __restrict__ kkbias) {
  const int n = blockIdx.x * 128 + threadIdx.x;
  if (n >= NN_) return;
  const float wL = 0.70710678118f;
  const float wC = 0.23570226039f;            // sqrt(2/36)
  const float s1 = wL / 5.65685424949f;       // wL / sqrt(C)
  float Rm[9], tv[3];
#pragma unroll
  for (int i = 0; i < 9; ++i) Rm[i] = R[(size_t)n * 9 + i];
#pragma unroll
  for (int i = 0; i < 3; ++i) tv[i] = tvec[(size_t)n * 3 + i];
  for (int h = 0; h < H_; ++h) {
    const float hw = headw[h];
    const float gamma = (hw > 20.0f) ? hw : log1pf(__expf(hw));
    const float s2 = wL * gamma * wC;
    float* qc = qcat + ((size_t)h * NN_ + n) * 48;
    float* kc = kcat + ((size_t)h * NN_ + n) * 48;
    float* vc = vcat + ((size_t)h * NN_ + n) * 48;
    for (int c = 0; c < 32; ++c) {
      qc[c] = qb[(size_t)n * 128 + h * 32 + c] * s1;
      kc[c] = kb[(size_t)n * 128 + h * 32 + c];
      vc[c] = vb[(size_t)n * 128 + h * 32 + c];
    }
    float kk = 0.0f;
    for (int p = 0; p < 4; ++p) {
      const float* ql = qp + (size_t)n * 48 + (h * 4 + p) * 3;
      const float* kl = kp + (size_t)n * 48 + (h * 4 + p) * 3;
      const float* vl = vp + (size_t)n * 48 + (h * 4 + p) * 3;
#pragma unroll
      for (int i = 0; i < 3; ++i) {
        const float gq = Rm[i*3+0]*ql[0] + Rm[i*3+1]*ql[1] + Rm[i*3+2]*ql[2] + tv[i];
        const float gk = Rm[i*3+0]*kl[0] + Rm[i*3+1]*kl[1] + Rm[i*3+2]*kl[2] + tv[i];
        const float gv = Rm[i*3+0]*vl[0] + Rm[i*3+1]*vl[1] + Rm[i*3+2]*vl[2] + tv[i];
        qc[32 + p * 3 + i] = gq * s2;
        kc[32 + p * 3 + i] = gk;
        vc[32 + p * 3 + i] = gv;
        kk += gk * gk;
      }
    }
    for (int c = 44; c < 48; ++c) { qc[c] = 0.0f; kc[c] = 0.0f; vc[c] = 0.0f; }
    kkbias[(size_t)h * NN_ + n] = -0.5f * s2 * kk;
  }
}

// ---------------------------------------------------------------------------
// IPA finalize: feat[n] = [ o(128) , opt_local(48) , norms(16) ]  (192 cols)
// opt_local = R^T (opt_global - t);  norms = sqrt(|opt_local|^2 + 1e-8)
// ---------------------------------------------------------------------------
__global__ __launch_bounds__(128)
void ipa_finalize(const float* __restrict__ ocat, const float* __restrict__ R,
                  const float* __restrict__ tvec, float* __restrict__ feat) {
  const int n = blockIdx.x * 128 + threadIdx.x;
  if (n >= NN_) return;
  float Rm[9], tv[3];
#pragma unroll
  for (int i = 0; i < 9; ++i) Rm[i] = R[(size_t)n * 9 + i];
#pragma unroll
  for (int i = 0; i < 3; ++i) tv[i] = tvec[(size_t)n * 3 + i];
  float* f = feat + (size_t)n * 192;
  for (int h = 0; h < H_; ++h) {
    const float* oc = ocat + ((size_t)h * NN_ + n) * 48;
    for (int c = 0; c < 32; ++c) f[h * 32 + c] = oc[c];
    for (int p = 0; p < 4; ++p) {
      const float d0 = oc[32 + p * 3 + 0] - tv[0];
      const float d1 = oc[32 + p * 3 + 1] - tv[1];
      const float d2 = oc[32 + p * 3 + 2] - tv[2];
      const float l0 = Rm[0] * d0 + Rm[3] * d1 + Rm[6] * d2;  // R^T
      const float l1 = Rm[1] * d0 + Rm[4] * d1 + Rm[7] * d2;
      const float l2 = Rm[2] * d0 + Rm[5] * d1 + Rm[8] * d2;
      f[128 + (h * 4 + p) * 3 + 0] = l0;
      f[128 + (h * 4 + p) * 3 + 1] = l1;
      f[128 + (h * 4 + p) * 3 + 2] = l2;
      f[176 + h * 4 + p] = sqrtf(l0 * l0 + l1 * l1 + l2 * l2 + 1e-8f);
    }
  }
}

// ---------------------------------------------------------------------------
// Host-side GEMM dispatch helpers.
// ---------------------------------------------------------------------------
static inline void gemm(hipStream_t s, int blayout, bool gelu,
                        const float* A, int lda, const float* B, int ldb,
                        float* C, int ldc, int M, int Nc, int K,
                        const float* bias, const float* resid, int ldres,
                        float alpha) {
  dim3 g(Nc / 16, M / 16), blk(32);
  if (blayout == 1)
    gemm_wmma_f32<1, false><<<g, blk, 0, s>>>(A, lda, B, ldb, C, ldc, K, bias, resid, ldres, alpha);
  else if (gelu)
    gemm_wmma_f32<0, true><<<g, blk, 0, s>>>(A, lda, B, ldb, C, ldc, K, bias, resid, ldres, alpha);
  else
    gemm_wmma_f32<0, false><<<g, blk, 0, s>>>(A, lda, B, ldb, C, ldc, K, bias, resid, ldres, alpha);
}

static inline void gemm_sk(hipStream_t s,
                           const float* A, int lda, const float* B, int ldb,
                           float* C, int ldc, int M, int Nc, int K,
                           const float* bias, const float* resid, int ldres,
                           float alpha) {
  dim3 g(Nc / 16, M / 16), blk(256);
  gemm_wmma_f32_splitk<<<g, blk, 0, s>>>(A, lda, B, ldb, C, ldc, K, bias, resid, ldres, alpha);
}

extern "C" void kernel_launch(void* const* d_in, const int* in_sizes, int n_in,
                              void* d_out, int out_size, void* d_ws, size_t ws_size,
                              hipStream_t stream) {
  (void)in_sizes; (void)n_in; (void)out_size; (void)ws_size;
  const float* seq   = (const float*)d_in[0];
  const float* pair  = (const float*)d_in[1];
  const float* Rfrm  = (const float*)d_in[2];
  const float* tfrm  = (const float*)d_in[3];
  const float* ln1g  = (const float*)d_in[4];
  const float* ln1b  = (const float*)d_in[5];
  const float* wqkv  = (const float*)d_in[6];
  const float* bqkv  = (const float*)d_in[7];
  const float* wo    = (const float*)d_in[8];
  const float* bo    = (const float*)d_in[9];
  const float* w21   = (const float*)d_in[10];
  const float* b21   = (const float*)d_in[11];
  const float* lnffg = (const float*)d_in[12];
  const float* lnffb = (const float*)d_in[13];
  const float* wff1  = (const float*)d_in[14];
  const float* bff1  = (const float*)d_in[15];
  const float* wff2  = (const float*)d_in[16];
  const float* bff2  = (const float*)d_in[17];
  const float* wl    = (const float*)d_in[18];
  const float* bl    = (const float*)d_in[19];
  const float* wr    = (const float*)d_in[20];
  const float* br    = (const float*)d_in[21];
  const float* wconv = (const float*)d_in[22];
  const float* bconv = (const float*)d_in[23];
  const float* iwq   = (const float*)d_in[24];
  const float* iwk   = (const float*)d_in[25];
  const float* iwv   = (const float*)d_in[26];
  const float* iwqp  = (const float*)d_in[27];
  const float* iwkp  = (const float*)d_in[28];
  const float* iwvp  = (const float*)d_in[29];
  const float* headw = (const float*)d_in[30];
  const float* iwout = (const float*)d_in[31];
  const float* ibout = (const float*)d_in[32];
  const float* ln3g  = (const float*)d_in[33];
  const float* ln3b  = (const float*)d_in[34];

  float* out_x    = (float*)d_out;                       // (N, D1)
  float* out_pair = out_x + (size_t)NN_ * D1_;           // (N, N, D2)
  float* out_R    = out_pair + (size_t)NN_ * NN_ * D2_;  // (N, 3, 3)
  float* out_t    = out_R + (size_t)NN_ * 9;             // (N, 3)

  // Workspace allocator (floats).
  float* w = (float*)d_ws;
  size_t off = 0;
  auto alloc = [&](size_t n) { float* p = w + off; off += (n + 63) & ~(size_t)63; return p; };
  float* xn     = alloc((size_t)NN_ * D1_);
  float* qkv    = alloc((size_t)NN_ * 3 * D1_);
  float* logits = alloc((size_t)H_ * NN_ * NN_);   // reused: MHA then IPA
  float* abuf   = alloc((size_t)NN_ * D1_);
  float* x1     = alloc((size_t)NN_ * D1_);
  float* pmean  = alloc((size_t)NN_ * D2_);
  float* hdn    = alloc((size_t)NN_ * D1_);
  float* ffh    = alloc((size_t)NN_ * 4 * D1_);
  float* x2     = alloc((size_t)NN_ * D1_);
  float* lbuf   = alloc((size_t)NN_ * 32);
  float* rbuf   = alloc((size_t)NN_ * 32);
  float* qb     = alloc((size_t)NN_ * D1_);
  float* kb     = alloc((size_t)NN_ * D1_);
  float* vb     = alloc((size_t)NN_ * D1_);
  float* qpb    = alloc((size_t)NN_ * 48);
  float* kpb    = alloc((size_t)NN_ * 48);
  float* vpb    = alloc((size_t)NN_ * 48);
  float* qcat   = alloc((size_t)H_ * NN_ * 48);
  float* kcat   = alloc((size_t)H_ * NN_ * 48);
  float* vcat   = alloc((size_t)H_ * NN_ * 48);
  float* ocat   = alloc((size_t)H_ * NN_ * 48);
  float* kkb    = alloc((size_t)H_ * NN_);
  float* feat   = alloc((size_t)NN_ * 192);
  float* ipao   = alloc((size_t)NN_ * D1_);

  const float inv_sqrt_c = 0.17677669529f;  // 1/sqrt(32)

  // ---- 1D track: pre-LN MHA -------------------------------------------------
  layernorm128<<<NN_, 128, 0, stream>>>(seq, ln1g, ln1b, nullptr, xn);
  gemm(stream, 0, false, xn, 128, wqkv, 384, qkv, 384, NN_, 384, 128, bqkv, nullptr, 0, 1.0f);
  for (int h = 0; h < H_; ++h)
    gemm(stream, 1, false, qkv + h * 32, 384, qkv + 128 + h * 32, 384,
         logits + (size_t)h * NN_ * NN_, NN_, NN_, NN_, C_, nullptr, nullptr, 0, inv_sqrt_c);
  softmax_bias<<<H_ * NN_, 256, 0, stream>>>(logits, nullptr);
  for (int h = 0; h < H_; ++h)
    gemm_sk(stream, logits + (size_t)h * NN_ * NN_, NN_, qkv + 256 + h * 32, 384,
            abuf + h * 32, 128, NN_, 32, NN_, nullptr, nullptr, 0, 1.0f);
  // x1 = seq + attn@w_o + b_o
  gemm(stream, 0, false, abuf, 128, wo, 128, x1, 128, NN_, 128, 128, bo, seq, 128, 1.0f);

  // ---- pair -> 1D aggregation ----------------------------------------------
  pair_mean<<<NN_, 256, 0, stream>>>(pair, pmean);
  gemm(stream, 0, false, pmean, 64, w21, 128, x1, 128, NN_, 128, 64, b21, x1, 128, 1.0f);

  // ---- FFN ------------------------------------------------------------------
  layernorm128<<<NN_, 128, 0, stream>>>(x1, lnffg, lnffb, nullptr, hdn);
  gemm(stream, 0, true,  hdn, 128, wff1, 512, ffh, 512, NN_, 512, 128, bff1, nullptr, 0, 1.0f);
  gemm(stream, 0, false, ffh, 512, wff2, 128, x2, 128, NN_, 128, 512, bff2, x1, 128, 1.0f);

  // ---- pair update: outer-sum + 1x1 conv, streamed once to output ----------
  gemm(stream, 0, false, x2, 128, wl, 32, lbuf, 32, NN_, 32, 128, bl, nullptr, 0, 1.0f);
  gemm(stream, 0, false, x2, 128, wr, 32, rbuf, 32, NN_, 32, 128, br, nullptr, 0, 1.0f);
  pair_conv<<<dim3(NN_ / 32, NN_), 256, 0, stream>>>(pair, lbuf, rbuf, wconv, bconv, out_pair);

  // ---- IPA ------------------------------------------------------------------
  gemm(stream, 0, false, x2, 128, iwq,  128, qb,  128, NN_, 128, 128, nullptr, nullptr, 0, 1.0f);
  gemm(stream, 0, false, x2, 128, iwk,  128, kb,  128, NN_, 128, 128, nullptr, nullptr, 0, 1.0f);
  gemm(stream, 0, false, x2, 128, iwv,  128, vb,  128, NN_, 128, 128, nullptr, nullptr, 0, 1.0f);
  gemm(stream, 0, false, x2, 128, iwqp, 48,  qpb, 48,  NN_, 48,  128, nullptr, nullptr, 0, 1.0f);
  gemm(stream, 0, false, x2, 128, iwkp, 48,  kpb, 48,  NN_, 48,  128, nullptr, nullptr, 0, 1.0f);
  gemm(stream, 0, false, x2, 128, iwvp, 48,  vpb, 48,  NN_, 48,  128, nullptr, nullptr, 0, 1.0f);
  ipa_pointprep<<<NN_ / 128, 128, 0, stream>>>(qb, kb, vb, qpb, kpb, vpb, Rfrm, tfrm,
                                               headw, qcat, kcat, vcat, kkb);
  for (int h = 0; h < H_; ++h)
    gemm(stream, 1, false, qcat + (size_t)h * NN_ * 48, 48, kcat + (size_t)h * NN_ * 48, 48,
         logits + (size_t)h * NN_ * NN_, NN_, NN_, NN_, 48, nullptr, nullptr, 0, 1.0f);
  softmax_bias<<<H_ * NN_, 256, 0, stream>>>(logits, kkb);
  for (int h = 0; h < H_; ++h)
    gemm_sk(stream, logits + (size_t)h * NN_ * NN_, NN_, vcat + (size_t)h * NN_ * 48, 48,
            ocat + (size_t)h * NN_ * 48, 48, NN_, 48, NN_, nullptr, nullptr, 0, 1.0f);
  ipa_finalize<<<NN_ / 128, 128, 0, stream>>>(ocat, Rfrm, tfrm, feat);
  gemm(stream, 0, false, feat, 192, iwout, 128, ipao, 128, NN_, 128, 192, ibout, nullptr, 0, 1.0f);

  // ---- x_out = x2 + LN(ipa_out) --------------------------------------------
  layernorm128<<<NN_, 128, 0, stream>>>(ipao, ln3g, ln3b, x2, out_x);

  // ---- pass-through frame outputs ------------------------------------------
  (void)hipMemcpyAsync(out_R, Rfrm, (size_t)NN_ * 9 * sizeof(float), hipMemcpyDeviceToDevice, stream);
  (void)hipMemcpyAsync(out_t, tfrm, (size_t)NN_ * 3 * sizeof(float), hipMemcpyDeviceToDevice, stream);
}